// GwcVolume_85048942395524
// MI455X (gfx1250) — compile-verified
//
#include <hip/hip_runtime.h>

// GwcVolume (group-wise correlation cost volume), MI455X / gfx1250.
// Store-bound: fp32 end-to-end, V_WMMA_F32_16X16X4_F32 for the 8-deep channel
// dot products, async global->LDS staging, branchless LDS transpose, NT stores.

typedef __attribute__((ext_vector_type(2))) float v2f;
typedef __attribute__((ext_vector_type(8))) float v8f;
typedef int v4i __attribute__((vector_size(4 * sizeof(int))));

#define BB    2
#define CC    320
#define HH    128
#define WW    240
#define GG    40
#define CPG   8
#define MAXD  48
#define WPAD  (MAXD + WW)   // 288: R row padded with 48 leading zeros
#define NTIL  15            // WW / 16 tiles

#if __has_builtin(__builtin_amdgcn_global_load_async_to_lds_b128) && \
    __has_builtin(__builtin_amdgcn_s_wait_asynccnt)
#define USE_ASYNC_LDS 1
#else
#define USE_ASYNC_LDS 0
#endif

__global__ __launch_bounds__(256) void gwc_volume_kernel(
    const float* __restrict__ left,
    const float* __restrict__ right,
    float* __restrict__ out)
{
    __shared__ float Lrow[CPG][WW];      //  7680 B
    __shared__ float Rp[CPG][WPAD];      //  9216 B (zero-padded left by MAXD)
    __shared__ float Ob[MAXD + 1][WW];   // 47040 B (row 48 = write dump row)

    const int tid = threadIdx.x;
    const int bgh = blockIdx.x;              // over B*G*H = 10240
    const int b   = bgh / (GG * HH);
    const int gh  = bgh % (GG * HH);
    const int g   = gh / HH;
    const int h   = gh % HH;

    const size_t chanStride = (size_t)HH * WW;
    const float* lp = left  + ((size_t)(b * CC + g * CPG) * HH + h) * WW;
    const float* rp = right + ((size_t)(b * CC + g * CPG) * HH + h) * WW;

    // Zero the pad region of Rp (j < 0 -> exact zero-padding of the output).
    for (int i = tid; i < CPG * MAXD; i += 256) {
        Rp[i / MAXD][i % MAXD] = 0.0f;
    }

#if USE_ASYNC_LDS
    // CDNA5 async DMA path: GLOBAL_LOAD_ASYNC_TO_LDS_B128, tracked by ASYNCcnt.
    // 8 rows x 60 float4 per input = 480 chunks; all addresses 16B-aligned.
    {
        typedef __attribute__((address_space(1))) v4i glb_v4i;
        typedef __attribute__((address_space(3))) v4i lds_v4i;
        for (int i = tid; i < CPG * (WW / 4); i += 256) {
            const int c  = i / (WW / 4);
            const int w4 = (i % (WW / 4)) * 4;
            __builtin_amdgcn_global_load_async_to_lds_b128(
                (glb_v4i*)(lp + (size_t)c * chanStride + w4),
                (lds_v4i*)&Lrow[c][w4], 0, 0);
            __builtin_amdgcn_global_load_async_to_lds_b128(
                (glb_v4i*)(rp + (size_t)c * chanStride + w4),
                (lds_v4i*)&Rp[c][MAXD + w4], 0, 0);
        }
        __builtin_amdgcn_s_wait_asynccnt(0);
    }
#else
    for (int i = tid; i < CPG * WW; i += 256) {
        const int c = i / WW, w = i % WW;
        Lrow[c][w]      = __builtin_nontemporal_load(lp + (size_t)c * chanStride + w);
        Rp[c][MAXD + w] = __builtin_nontemporal_load(rp + (size_t)c * chanStride + w);
    }
#endif
    __syncthreads();

    const int lane = tid & 31;
    const int wave = tid >> 5;
    const int n    = lane & 15;          // column (w in tile) == row m for A
    const int c0   = (lane >> 4) * 2;    // K-half: VGPR0 <-> K={0|2}, VGPR1 <-> K+1
    const int mb   = (lane >> 4) * 8;    // accumulator row base

    // P[j,w] = sum_c R[c,j] * L[c,w];  out[d,w] = P[w-d, w] / 8.
    // Per 16-wide w-tile: 4 row tiles (j0 = w0-48+16k) cover d in [0,48) once each.
    for (int t = wave; t < NTIL; t += 8) {
        const int w0 = t * 16;

        v2f bv, b2;                       // B = L tile (4x16 per WMMA)
        bv.x = Lrow[c0][w0 + n];
        bv.y = Lrow[c0 + 1][w0 + n];
        b2.x = Lrow[c0 + 4][w0 + n];
        b2.y = Lrow[c0 + 5][w0 + n];

        #pragma unroll
        for (int kt = 0; kt < 4; ++kt) {
            const int xi = w0 + 16 * kt + n;   // Rp index = j0 + m + 48
            v2f av, a2;                        // A = R^T tile (16x4 per WMMA)
            av.x = Rp[c0][xi];
            av.y = Rp[c0 + 1][xi];
            a2.x = Rp[c0 + 4][xi];
            a2.y = Rp[c0 + 5][xi];

            v8f acc = {};
            acc = __builtin_amdgcn_wmma_f32_16x16x4_f32(
                false, av, false, bv, (short)0, acc, false, false);
            acc = __builtin_amdgcn_wmma_f32_16x16x4_f32(
                false, a2, false, b2, (short)0, acc, false, false);

            const int dbase = n - mb + 48 - 16 * kt;
            #pragma unroll
            for (int r = 0; r < 8; ++r) {
                const int d = dbase - r;       // d = n - m + 48 - 16k
                int row;
                if (kt == 0 || kt == 3) {
                    // Boundary tiles: clamp invalid d to dump row 48 (branchless).
                    row = ((unsigned)d < (unsigned)MAXD) ? d : MAXD;
                } else {
                    row = d;                   // kt=1: d in [17,47]; kt=2: d in [1,31]
                }
                Ob[row][w0 + n] = 0.125f * acc[r];   // mean over cpg = 8
            }
        }
    }
    __syncthreads();

    // Coalesced streaming writeout: rows of 240 contiguous fp32 (960 B), NT.
    float* op = out + (size_t)(b * GG + g) * MAXD * chanStride + (size_t)h * WW;
    for (int i = tid; i < MAXD * WW; i += 256) {
        const int d = i / WW, w = i % WW;
        __builtin_nontemporal_store(Ob[d][w], op + (size_t)d * chanStride + w);
    }
}

extern "C" void kernel_launch(void* const* d_in, const int* in_sizes, int n_in,
                              void* d_out, int out_size, void* d_ws, size_t ws_size,
                              hipStream_t stream) {
    const float* left  = (const float*)d_in[0];
    const float* right = (const float*)d_in[1];
    float*       out   = (float*)d_out;
    (void)in_sizes; (void)n_in; (void)out_size; (void)d_ws; (void)ws_size;

    dim3 grid(BB * GG * HH);   // 10240 workgroups, one per (b, g, h)
    gwc_volume_kernel<<<grid, 256, 0, stream>>>(left, right, out);
}